// MultiHeadAdditiveAttentionBlockQ_69020124447492
// MI455X (gfx1250) — compile-verified
//
#include <hip/hip_runtime.h>
#include <hip/hip_bf16.h>

typedef __attribute__((ext_vector_type(2))) float v2f;
typedef __attribute__((ext_vector_type(8))) float v8f;

// V_WMMA_F32_16X16X4_F32: D = A(16x4 f32) x B(4x16 f32) + C(16x16 f32)
#define WMMA_F32(a, b, c) \
  __builtin_amdgcn_wmma_f32_16x16x4_f32(false, (a), false, (b), (short)0, (c), false, false)

constexpr int Bn = 32, T = 256, TM1 = 255, Dm = 15, Hh = 3, KV = 5, QD = 9, FF = 75;

// ---------------------------------------------------------------------------
// Kernel 1: one block (8 waves) per (batch, query-token).
//   - build 255 x 5 geometric key features into LDS (padded 256 x 8)
//   - K/V projections (256x8 @ 8x16) with v_wmma_f32_16x16x4_f32
//   - additive-attention scores; softmax via wave32 shfl butterflies +
//     one cross-wave LDS combine; two-level parallel weighted value sum
//   - writes x_att (b,t,15) into `xout` (d_out reused as intermediate)
// ---------------------------------------------------------------------------
__global__ __launch_bounds__(256) void attn_kernel(
    const float* __restrict__ state, const float* __restrict__ action,
    const float* __restrict__ Wk, const float* __restrict__ Wq,
    const float* __restrict__ Wv, const float* __restrict__ att_bias,
    const float* __restrict__ Wscore, const float* __restrict__ bscore,
    float* __restrict__ xout) {
  __shared__ float kv_lds[256][8];    // padded key features (K 5->8, rows 255->256)
  __shared__ float key_lds[256][16];  // key projections (N 15->16)
  __shared__ float val_lds[256][16];  // value projections
  __shared__ float red[256][3];       // per-key softmax weights
  __shared__ float wred[8][3];        // cross-wave reduction scratch
  __shared__ float pres[16][16];      // weighted-sum partials [group][col]
  __shared__ float qproj[16];         // query projection (15)

  const int tid = threadIdx.x;
  const int bid = blockIdx.x;
  const int bq = bid / T;
  const int tq = bid % T;

  const float* sq = state + (bq * T + tq) * 7;
  const float si0 = sq[0], si1 = sq[1], si2 = sq[2], si3 = sq[3];

  // ---- per-key geometric features (rotate rel. offsets into key frame) ----
  if (tid < TM1) {
    const int key = tid + (tid >= tq);
    const float* sk = state + (bq * T + key) * 7;
    const float dx1 = sk[0] - si0, dy1 = sk[1] - si1;
    const float dx2 = sk[2] - si2, dy2 = sk[3] - si3;
    const float ang = -sk[6] + 1.57079632679489662f;
    const float c = cosf(ang), s = sinf(ang);
    const float rx1 = dx1 * c - dy1 * s, ry1 = dx1 * s + dy1 * c;
    const float rx2 = dx2 * c - dy2 * s, ry2 = dx2 * s + dy2 * c;
    const float r = sqrtf(rx1 * rx1 + ry1 * ry1);
    const float rt = 1.0f / (1.0f + expf(-1.0f + 5.0f * (r - 0.2f)));
    kv_lds[tid][0] = rx1 / r;
    kv_lds[tid][1] = ry1 / r;
    kv_lds[tid][2] = rx2;
    kv_lds[tid][3] = ry2;
    kv_lds[tid][4] = rt;
    kv_lds[tid][5] = 0.f;
    kv_lds[tid][6] = 0.f;
    kv_lds[tid][7] = 0.f;
  } else {  // tid == 255: zero padding row
    for (int d = 0; d < 8; ++d) kv_lds[tid][d] = 0.f;
  }

  // ---- query projection: q_x(9) @ Wq.T -> 15 ----
  if (tid < Dm) {
    float qx[QD];
    for (int d = 0; d < 7; ++d) qx[d] = sq[d];
    qx[7] = action[(bq * T + tq) * 2 + 0];
    qx[8] = action[(bq * T + tq) * 2 + 1];
    float acc = 0.f;
    for (int k = 0; k < QD; ++k) acc += qx[k] * Wq[tid * QD + k];
    qproj[tid] = acc;
  }
  __syncthreads();

  // ---- WMMA K/V projections: kv(256x8) @ W.T(8x16), fp32 16x16x4 tiles ----
  const int lane = tid & 31;
  const int wv = tid >> 5;       // wave id 0..7
  const int hi = lane >> 4;      // lane group (selects K pair / C row half)
  const int nn = lane & 15;      // column index

  v2f bk[2], bv[2];  // B operands for the two K-steps (rows 0-3 / 4-7 of W.T)
  for (int kk = 0; kk < 2; ++kk) {
    const int k0 = 4 * kk + 2 * hi;
    bk[kk].x = (k0 < KV && nn < Dm) ? Wk[nn * KV + k0] : 0.f;
    bk[kk].y = (k0 + 1 < KV && nn < Dm) ? Wk[nn * KV + k0 + 1] : 0.f;
    bv[kk].x = (k0 < KV && nn < Dm) ? Wv[nn * KV + k0] : 0.f;
    bv[kk].y = (k0 + 1 < KV && nn < Dm) ? Wv[nn * KV + k0 + 1] : 0.f;
  }
  for (int tile = 0; tile < 2; ++tile) {
    const int mt = 2 * wv + tile;           // M-tile of 16 key rows
    const int row = 16 * mt + nn;           // A row for this lane
    v8f ck = {};
    v8f cv = {};
    for (int kk = 0; kk < 2; ++kk) {
      const int k0 = 4 * kk + 2 * hi;
      v2f a;
      a.x = kv_lds[row][k0];
      a.y = kv_lds[row][k0 + 1];
      ck = WMMA_F32(a, bk[kk], ck);
      cv = WMMA_F32(a, bv[kk], cv);
    }
    for (int v = 0; v < 8; ++v) {  // C/D layout: row = v + 8*hi, col = nn
      key_lds[16 * mt + v + 8 * hi][nn] = ck[v];
      val_lds[16 * mt + v + 8 * hi][nn] = cv[v];
    }
  }
  __syncthreads();

  // ---- additive attention scores ----
  float ab[KV], ws[KV];
  for (int k = 0; k < KV; ++k) {
    ab[k] = att_bias[k];
    ws[k] = Wscore[k];
  }
  const float bs = bscore[0];

  float sc[Hh];
  if (tid < TM1) {
    for (int h = 0; h < Hh; ++h) {
      float acc = bs;
      for (int k = 0; k < KV; ++k) {
        const float e = tanhf(key_lds[tid][h * KV + k] + qproj[h * KV + k] + ab[k]);
        acc += ws[k] * e;
      }
      sc[h] = acc;
    }
  } else {
    sc[0] = sc[1] = sc[2] = -1e30f;
  }

  // ---- softmax: wave32 shfl_xor butterfly max, cross-wave combine ----
  float vmax[Hh];
  for (int h = 0; h < Hh; ++h) vmax[h] = sc[h];
  for (int m = 16; m > 0; m >>= 1)
    for (int h = 0; h < Hh; ++h)
      vmax[h] = fmaxf(vmax[h], __shfl_xor(vmax[h], m, 32));
  if (lane == 0)
    for (int h = 0; h < Hh; ++h) wred[wv][h] = vmax[h];
  __syncthreads();
  float mx[Hh];
  for (int h = 0; h < Hh; ++h) {
    float m = wred[0][h];
    for (int g = 1; g < 8; ++g) m = fmaxf(m, wred[g][h]);
    mx[h] = m;
  }
  __syncthreads();  // all reads of wred done before it is overwritten

  float ex[Hh], vsum[Hh];
  for (int h = 0; h < Hh; ++h) {
    ex[h] = (tid < TM1) ? expf(sc[h] - mx[h]) : 0.f;
    vsum[h] = ex[h];
  }
  for (int m = 16; m > 0; m >>= 1)
    for (int h = 0; h < Hh; ++h) vsum[h] += __shfl_xor(vsum[h], m, 32);
  if (lane == 0)
    for (int h = 0; h < Hh; ++h) wred[wv][h] = vsum[h];
  __syncthreads();
  float sm[Hh];
  for (int h = 0; h < Hh; ++h) {
    float s = wred[0][h];
    for (int g = 1; g < 8; ++g) s += wred[g][h];
    sm[h] = s;
  }
  for (int h = 0; h < Hh; ++h) red[tid][h] = ex[h] / sm[h];
  __syncthreads();

  // ---- weighted value sum, two-level parallel ----
  // level 1: thread (g = tid>>4, n = tid&15) sums 16 keys for column n.
  {
    const int g = tid >> 4;
    const int n = tid & 15;
    const int h = (n < Dm) ? (n / KV) : 0;  // col 15 is zero-padded anyway
    float acc = 0.f;
    for (int i = 0; i < 16; ++i) {
      const int j = g * 16 + i;  // j==255 has weight 0 and zero value row
      acc += red[j][h] * val_lds[j][n];
    }
    pres[g][n] = acc;
  }
  __syncthreads();
  // level 2: 15 lanes combine 16 partials per column and store.
  if (tid < Dm) {
    float acc = 0.f;
    for (int g = 0; g < 16; ++g) acc += pres[g][tid];
    xout[(bq * T + tq) * Dm + tid] = acc;
  }
}

// ---------------------------------------------------------------------------
// Kernel 2: one wave per 16 tokens. LN1 -> FFN (WMMA fp32) -> residual -> LN2.
// Reads x_att from `xbuf` (written by kernel 1) and overwrites it in place.
// ---------------------------------------------------------------------------
__global__ __launch_bounds__(32) void ffn_kernel(
    float* xbuf, const float* __restrict__ g1, const float* __restrict__ beta1,
    const float* __restrict__ W1, const float* __restrict__ bff1,
    const float* __restrict__ W2, const float* __restrict__ bff2,
    const float* __restrict__ g2, const float* __restrict__ beta2) {
  __shared__ float X[16][16];  // LN1 output (K 15->16 padded)
  __shared__ float Y[16][80];  // hidden relu(x@W1.T+b) (75->80 padded)
  __shared__ float R[16][16];  // residual result

  const int lane = threadIdx.x;
  const int hi = lane >> 4;
  const int m = lane & 15;
  const int base = blockIdx.x * 16;

  // ---- LN1 (one token per low lane) ----
  if (lane < 16) {
    const float* xp = xbuf + (base + lane) * Dm;
    float x[Dm];
    float mean = 0.f;
    for (int d = 0; d < Dm; ++d) {
      x[d] = xp[d];
      mean += x[d];
    }
    mean *= (1.f / Dm);
    float var = 0.f;
    for (int d = 0; d < Dm; ++d) {
      const float t = x[d] - mean;
      var += t * t;
    }
    var *= (1.f / Dm);
    const float rs = rsqrtf(var + 1e-5f);
    for (int d = 0; d < Dm; ++d) X[lane][d] = (x[d] - mean) * rs * g1[d] + beta1[d];
    X[lane][15] = 0.f;
  }
  __syncthreads();

  // ---- FFN1: X(16x16) @ W1.T(16x80) -> Y, relu+bias. 5 N-tiles x 4 K-steps.
  for (int nt = 0; nt < 5; ++nt) {
    v8f acc = {};
    const int col = 16 * nt + m;
    for (int kk = 0; kk < 4; ++kk) {
      const int k0 = 4 * kk + 2 * hi;
      v2f a;
      a.x = X[m][k0];
      a.y = X[m][k0 + 1];
      v2f b;
      b.x = (k0 < Dm && col < FF) ? W1[col * Dm + k0] : 0.f;
      b.y = (k0 + 1 < Dm && col < FF) ? W1[col * Dm + k0 + 1] : 0.f;
      acc = WMMA_F32(a, b, acc);
    }
    const float bias = (col < FF) ? bff1[col] : 0.f;
    for (int v = 0; v < 8; ++v) {
      const int row = v + 8 * hi;
      Y[row][16 * nt + m] = (col < FF) ? fmaxf(acc[v] + bias, 0.f) : 0.f;
    }
  }
  __syncthreads();

  // ---- FFN2: Y(16x80) @ W2.T(80x16) -> 16x16. 20 K-steps. ----
  v8f acc2 = {};
  for (int kk = 0; kk < 20; ++kk) {
    const int k0 = 4 * kk + 2 * hi;
    v2f a;
    a.x = Y[m][k0];
    a.y = Y[m][k0 + 1];
    v2f b;
    b.x = (k0 < FF && m < Dm) ? W2[m * FF + k0] : 0.f;
    b.y = (k0 + 1 < FF && m < Dm) ? W2[m * FF + k0 + 1] : 0.f;
    acc2 = WMMA_F32(a, b, acc2);
  }
  const float b2 = (m < Dm) ? bff2[m] : 0.f;
  for (int v = 0; v < 8; ++v) {
    const int row = v + 8 * hi;
    R[row][m] = acc2[v] + b2 + X[row][m];  // residual with LN1 output
  }
  __syncthreads();

  // ---- LN2 and final store ----
  if (lane < 16) {
    float x[Dm];
    float mean = 0.f;
    for (int d = 0; d < Dm; ++d) {
      x[d] = R[lane][d];
      mean += x[d];
    }
    mean *= (1.f / Dm);
    float var = 0.f;
    for (int d = 0; d < Dm; ++d) {
      const float t = x[d] - mean;
      var += t * t;
    }
    var *= (1.f / Dm);
    const float rs = rsqrtf(var + 1e-5f);
    float* op = xbuf + (base + lane) * Dm;
    for (int d = 0; d < Dm; ++d) op[d] = (x[d] - mean) * rs * g2[d] + beta2[d];
  }
}

extern "C" void kernel_launch(void* const* d_in, const int* in_sizes, int n_in,
                              void* d_out, int out_size, void* d_ws, size_t ws_size,
                              hipStream_t stream) {
  (void)in_sizes; (void)n_in; (void)out_size; (void)d_ws; (void)ws_size;
  const float* state  = (const float*)d_in[0];
  const float* action = (const float*)d_in[1];
  const float* Wk     = (const float*)d_in[2];
  const float* Wq     = (const float*)d_in[3];
  const float* Wv     = (const float*)d_in[4];
  const float* ab     = (const float*)d_in[5];
  const float* Wsc    = (const float*)d_in[6];
  const float* bsc    = (const float*)d_in[7];
  const float* g1     = (const float*)d_in[8];
  const float* beta1  = (const float*)d_in[9];
  const float* W1     = (const float*)d_in[10];
  const float* bff1   = (const float*)d_in[11];
  const float* W2     = (const float*)d_in[12];
  const float* bff2   = (const float*)d_in[13];
  const float* g2     = (const float*)d_in[14];
  const float* beta2  = (const float*)d_in[15];
  float* out = (float*)d_out;

  attn_kernel<<<Bn * T, 256, 0, stream>>>(state, action, Wk, Wq, Wv, ab, Wsc, bsc, out);
  ffn_kernel<<<(Bn * T) / 16, 32, 0, stream>>>(out, g1, beta1, W1, bff1, W2, bff2, g2, beta2);
}